// UncertaintyWeightedTAL_21517786153378
// MI455X (gfx1250) — compile-verified
//
#include <hip/hip_runtime.h>
#include <math.h>

#define B_N 4096
#define D_K 512
#define TAU 0.02f
#define INV_TAU 50.0f
#define CLAMP_MAXV 1e36f

typedef float v2f __attribute__((ext_vector_type(2)));
typedef float v8f __attribute__((ext_vector_type(8)));

// ---------------------------------------------------------------------------
// K0: reciprocal L2 norms of each row. One wave (32 lanes) per row.
// rows [0, B_N)      -> textual  -> rnt
// rows [B_N, 2*B_N)  -> visual   -> rnv
// ---------------------------------------------------------------------------
__global__ void __launch_bounds__(256)
row_rnorm_kernel(const float* __restrict__ vis,
                 const float* __restrict__ txt,
                 float* __restrict__ rnv,
                 float* __restrict__ rnt)
{
    const int gwave = (int)((blockIdx.x * blockDim.x + threadIdx.x) >> 5);
    const int lane  = (int)(threadIdx.x & 31);

    const float* src;
    float* dst;
    int row;
    if (gwave < B_N) { src = txt; dst = rnt; row = gwave; }
    else             { src = vis; dst = rnv; row = gwave - B_N; }

    const float* p = src + (size_t)row * D_K;
    float s = 0.0f;
    #pragma unroll 4
    for (int j = lane; j < D_K; j += 32) {
        float x = p[j];
        s += x * x;
    }
    #pragma unroll
    for (int off = 16; off > 0; off >>= 1)
        s += __shfl_xor(s, off, 32);

    if (lane == 0)
        dst[row] = 1.0f / fmaxf(sqrtf(s), 1e-12f);
}

// ---------------------------------------------------------------------------
// K1: scores = (tn @ vn^T) with deferred normalization, f32 WMMA 16x16x4.
//
// Block = 256 threads = 8 waves. Block tile = 16 (M) x 512 (N); each wave
// computes a 16x64 strip = 4 accumulators of 16x16.
//
// ISA fragment layout (32-bit, 16x4 A / 4x16 B):
//   lane<16 : M/N = lane,    K = {0,1} in the two VGPRs
//   lane>=16: M/N = lane-16, K = {2,3}
// Both operands are row-major with K contiguous (NT GEMM), so each lane
// loads a float2 at [row_or_col * 512 + k + 2*(lane>=16)].
//
// C/D layout: VGPR r, lanes 0-15 -> M = r, lanes 16-31 -> M = r + 8; N = lane&15.
// ---------------------------------------------------------------------------
__global__ void __launch_bounds__(256)
scores_wmma_kernel(const float* __restrict__ txt,   // A rows (M x K)
                   const float* __restrict__ vis,   // B rows (N x K)
                   const float* __restrict__ rnt,
                   const float* __restrict__ rnv,
                   float* __restrict__ scores)
{
    const int lane   = (int)(threadIdx.x & 31);
    const int wave   = (int)(threadIdx.x >> 5);
    const int laneLo = lane & 15;
    const int laneHi = lane >> 4;

    const int bm = (int)blockIdx.x * 16;                 // M tile base
    const int bn = (int)blockIdx.y * 512 + wave * 64;    // N strip base (this wave)

    const float* aptr  = txt + (size_t)(bm + laneLo) * D_K + laneHi * 2;
    const float* bptr0 = vis + (size_t)(bn +  0 + laneLo) * D_K + laneHi * 2;
    const float* bptr1 = vis + (size_t)(bn + 16 + laneLo) * D_K + laneHi * 2;
    const float* bptr2 = vis + (size_t)(bn + 32 + laneLo) * D_K + laneHi * 2;
    const float* bptr3 = vis + (size_t)(bn + 48 + laneLo) * D_K + laneHi * 2;

    v8f acc0 = {}; v8f acc1 = {}; v8f acc2 = {}; v8f acc3 = {};

    #pragma unroll 4
    for (int k = 0; k < D_K; k += 4) {
        v2f a  = *(const v2f*)(aptr  + k);
        v2f b0 = *(const v2f*)(bptr0 + k);
        v2f b1 = *(const v2f*)(bptr1 + k);
        v2f b2 = *(const v2f*)(bptr2 + k);
        v2f b3 = *(const v2f*)(bptr3 + k);
        acc0 = __builtin_amdgcn_wmma_f32_16x16x4_f32(false, a, false, b0, (short)0, acc0, false, false);
        acc1 = __builtin_amdgcn_wmma_f32_16x16x4_f32(false, a, false, b1, (short)0, acc1, false, false);
        acc2 = __builtin_amdgcn_wmma_f32_16x16x4_f32(false, a, false, b2, (short)0, acc2, false, false);
        acc3 = __builtin_amdgcn_wmma_f32_16x16x4_f32(false, a, false, b3, (short)0, acc3, false, false);
    }

    // Deferred normalization scale factors.
    float rt[8];
    #pragma unroll
    for (int r = 0; r < 8; ++r)
        rt[r] = rnt[bm + laneHi * 8 + r];

    const int c0 = bn +  0 + laneLo;
    const int c1 = bn + 16 + laneLo;
    const int c2 = bn + 32 + laneLo;
    const int c3 = bn + 48 + laneLo;
    const float sv0 = rnv[c0];
    const float sv1 = rnv[c1];
    const float sv2 = rnv[c2];
    const float sv3 = rnv[c3];

    #pragma unroll
    for (int r = 0; r < 8; ++r) {
        const size_t rowoff = (size_t)(bm + laneHi * 8 + r) * B_N;
        scores[rowoff + c0] = acc0[r] * rt[r] * sv0;
        scores[rowoff + c1] = acc1[r] * rt[r] * sv1;
        scores[rowoff + c2] = acc2[r] * rt[r] * sv2;
        scores[rowoff + c3] = acc3[r] * rt[r] * sv3;
    }
}

// ---------------------------------------------------------------------------
// K2: per-row loss. One 256-thread block per row i.
// ---------------------------------------------------------------------------
__global__ void __launch_bounds__(256)
loss_rows_kernel(const float* __restrict__ scores,
                 const int* __restrict__ pid,
                 const float* __restrict__ margins,
                 float* __restrict__ per_losses)
{
    const int i   = (int)blockIdx.x;
    const int tid = (int)threadIdx.x;
    const int pid0 = pid[0];
    const float* srow = scores + (size_t)i * B_N;

    float epos = 0.0f, eposs = 0.0f, eneg = 0.0f, nneg = 0.0f;
    for (int j = tid; j < B_N; j += 256) {
        const float sc = srow[j];
        const float e  = expf(sc * INV_TAU);
        if (pid[j] != pid0) { eneg += e; nneg += 1.0f; }
        else                { epos += e; eposs += e * sc; }
    }

    #pragma unroll
    for (int off = 16; off > 0; off >>= 1) {
        epos  += __shfl_xor(epos,  off, 32);
        eposs += __shfl_xor(eposs, off, 32);
        eneg  += __shfl_xor(eneg,  off, 32);
        nneg  += __shfl_xor(nneg,  off, 32);
    }

    __shared__ float red[8][4];
    const int w = tid >> 5, lane = tid & 31;
    if (lane == 0) { red[w][0] = epos; red[w][1] = eposs; red[w][2] = eneg; red[w][3] = nneg; }
    __syncthreads();

    if (tid == 0) {
        float E = 0.0f, ES = 0.0f, EN = 0.0f, NN = 0.0f;
        #pragma unroll
        for (int q = 0; q < 8; ++q) { E += red[q][0]; ES += red[q][1]; EN += red[q][2]; NN += red[q][3]; }

        const float m   = margins[i];
        const float dot = ES / E;                         // sum(alpha * scores)
        const float neg_sum = fminf(EN, CLAMP_MAXV);
        const float loss_i2t = fmaxf(-dot + TAU * logf(neg_sum) + m, 0.0f);

        const float s0 = srow[0];
        const float t2i_sum = fminf(expf(s0 * INV_TAU) * NN, CLAMP_MAXV);
        const float loss_t2i = fmaxf(-s0 + TAU * logf(t2i_sum) + m, 0.0f);

        per_losses[i] = loss_i2t + loss_t2i;
    }
}

// ---------------------------------------------------------------------------
// K3: uncertainty weights + final weighted mean. Single 1024-thread block.
// ---------------------------------------------------------------------------
__global__ void __launch_bounds__(1024)
final_kernel(const float* __restrict__ per_losses,
             const float* __restrict__ unc,
             const int* __restrict__ stype,
             float* __restrict__ weights,
             float* __restrict__ final_loss)
{
    const int tid = (int)threadIdx.x;
    float sw = 0.0f, swl = 0.0f;
    for (int j = tid; j < B_N; j += 1024) {
        float w = 1.0f;
        if (stype[j] == 2)
            w = fminf(fmaxf(1.0f - 0.5f * unc[j], 0.1f), 1.0f);
        weights[j] = w;
        sw  += w;
        swl += w * per_losses[j];
    }

    #pragma unroll
    for (int off = 16; off > 0; off >>= 1) {
        sw  += __shfl_xor(sw,  off, 32);
        swl += __shfl_xor(swl, off, 32);
    }

    __shared__ float redW[32];
    __shared__ float redL[32];
    const int w = tid >> 5, lane = tid & 31;
    if (lane == 0) { redW[w] = sw; redL[w] = swl; }
    __syncthreads();

    if (tid < 32) {
        float a = redW[tid];
        float b = redL[tid];
        #pragma unroll
        for (int off = 16; off > 0; off >>= 1) {
            a += __shfl_xor(a, off, 32);
            b += __shfl_xor(b, off, 32);
        }
        if (tid == 0)
            final_loss[0] = b / a;
    }
}

// ---------------------------------------------------------------------------
// Launcher
// ---------------------------------------------------------------------------
extern "C" void kernel_launch(void* const* d_in, const int* in_sizes, int n_in,
                              void* d_out, int out_size, void* d_ws, size_t ws_size,
                              hipStream_t stream)
{
    (void)in_sizes; (void)n_in; (void)out_size; (void)ws_size;

    const float* vis     = (const float*)d_in[0];   // visual_feats  (B, D)
    const float* txt     = (const float*)d_in[1];   // textual_feats (B, D)
    const int*   pid     = (const int*)  d_in[2];   // (B,)
    const float* margins = (const float*)d_in[3];   // (B,)
    const float* unc     = (const float*)d_in[4];   // (B,)
    const int*   stype   = (const int*)  d_in[5];   // (B,)

    float* out        = (float*)d_out;
    float* final_loss = out;                 // [1]
    float* per_losses = out + 1;             // [B]
    float* weights    = out + 1 + B_N;       // [B]
    float* scores     = out + 1 + 2 * B_N;   // [B, B]

    float* rnt = (float*)d_ws;               // reciprocal norms, textual
    float* rnv = rnt + B_N;                  // reciprocal norms, visual

    // K0: 2*B rows, one wave per row, 8 waves per block.
    row_rnorm_kernel<<<dim3((2 * B_N) / 8), 256, 0, stream>>>(vis, txt, rnv, rnt);

    // K1: WMMA GEMM. Grid: (M tiles of 16) x (N strips of 512).
    scores_wmma_kernel<<<dim3(B_N / 16, B_N / 512), 256, 0, stream>>>(
        txt, vis, rnt, rnv, scores);

    // K2: per-row losses.
    loss_rows_kernel<<<dim3(B_N), 256, 0, stream>>>(scores, pid, margins, per_losses);

    // K3: weights + final scalar.
    final_kernel<<<dim3(1), 1024, 0, stream>>>(per_losses, unc, stype, weights, final_loss);
}